// BiRGAT_51634096833010
// MI455X (gfx1250) — compile-verified
//
#include <hip/hip_runtime.h>

// ---------------------------------------------------------------------------
// BiRGAT forward for MI455X (gfx1250, wave32).
// Dense GEMMs -> V_WMMA_F32_16X16X4_F32 (fp32-exact matrix pipe), branch-free
// inner loop (all GEMM shapes here have N%16==0, K%4==0; M tail via clamped
// loads + guarded stores). One wave computes a 64x16 tile (4 accumulators)
// so each strided B fragment feeds 4 WMMAs.
// Edge segment-softmax -> 3-pass f32 atomics (max / exp+sum / scatter);
// working set ~25MB fits in 192MB L2 so atomics resolve on-chip.
// ---------------------------------------------------------------------------

static constexpr int kSamp  = 2000;
static constexpr int kGenes = 10000;
static constexpr int kE     = 250000;
static constexpr int kInDim = 2000;
static constexpr int kP     = 64;     // projected dim (also H0, H1)
static constexpr int kH     = 64;     // per-head channels
static constexpr int kHC    = 128;    // HEADS * H
static constexpr int kNC    = 5;

typedef __attribute__((ext_vector_type(2))) float v2f;
typedef __attribute__((ext_vector_type(8))) float v8f;

#define GF_RELU 1
#define GF_BIAS 2
#define GF_ACC  4

// ---------------------------------------------------------------------------
// fp32 WMMA GEMM: C = [relu]( [C +] A[M,K] @ B[K,N] [+ bias] )
// Requirements (all calls in this file satisfy them): N % 16 == 0, K % 4 == 0.
// One wave owns MBLK vertically-adjacent 16x16 tiles (same 16 columns).
// Tile selection is wave-uniform, so EXEC is all ones at every v_wmma.
// ---------------------------------------------------------------------------
template <int FLAGS, int MBLK>
__global__ void gemm_wmma(const float* __restrict__ A, const float* __restrict__ B,
                          const float* __restrict__ bias, float* __restrict__ C,
                          int M, int N, int K, int total_sets, int ntn) {
  int set = blockIdx.x * blockDim.y + threadIdx.y;
  if (set >= total_sets) return;                  // wave-uniform exit
  int sm = (set / ntn) * (16 * MBLK);             // first output row of this set
  int tn = (set % ntn) * 16;
  int lane = threadIdx.x;                         // 0..31
  int l15 = lane & 15, half = lane >> 4;
  int n = tn + l15;                               // B/C column for this lane

  // clamped A rows (branchless tail handling; garbage rows never stored)
  int arow[MBLK];
#pragma unroll
  for (int t = 0; t < MBLK; ++t) {
    int m = sm + t * 16 + l15;
    arow[t] = m < M ? m : M - 1;
  }

  v8f zero = {};
  v8f acc[MBLK];
#pragma unroll
  for (int t = 0; t < MBLK; ++t) {
    if (FLAGS & GF_ACC) {
#pragma unroll
      for (int v = 0; v < 8; ++v) {
        int row = sm + t * 16 + v + 8 * half;
        row = row < M ? row : M - 1;
        acc[t][v] = C[row * N + n];
      }
    } else {
      acc[t] = zero;
    }
  }

  for (int k = 0; k < K; k += 4) {
    int ka = k + 2 * half;                        // lanes 0-15: K={k,k+1}; 16-31: K={k+2,k+3}
    v2f b;
    b.x = B[ka * N + n];
    b.y = B[(ka + 1) * N + n];
#pragma unroll
    for (int t = 0; t < MBLK; ++t) {
      v2f a;
      a.x = A[arow[t] * K + ka];                  // contiguous pair -> b64 load
      a.y = A[arow[t] * K + ka + 1];
      acc[t] = __builtin_amdgcn_wmma_f32_16x16x4_f32(false, a, false, b,
                                                     (short)0, acc[t], false, false);
    }
  }

  float bv = (FLAGS & GF_BIAS) ? bias[n] : 0.0f;
#pragma unroll
  for (int t = 0; t < MBLK; ++t) {
#pragma unroll
    for (int v = 0; v < 8; ++v) {
      int row = sm + t * 16 + v + 8 * half;
      if (row < M) {
        float x = acc[t][v] + bv;
        if (FLAGS & GF_RELU) x = x > 0.0f ? x : 0.0f;
        C[row * N + n] = x;
      }
    }
  }
}

// ---------------------------------------------------------------------------
__global__ void fill_f32(float* __restrict__ p, float v, int n) {
  int i = blockIdx.x * 256 + threadIdx.x;
  if (i < n) p[i] = v;
}

__device__ __forceinline__ void atomicMaxF32(float* addr, float val) {
  int v = __float_as_int(val);
  if (v >= 0) atomicMax((int*)addr, v);
  else        atomicMin((unsigned int*)addr, (unsigned int)v);
}

// pass 1: per-edge attention logits (leaky-relu + dot with att) and segment max
__global__ void edge_logits_max(const float* __restrict__ xl, const float* __restrict__ xr,
                                const int* __restrict__ src, const int* __restrict__ dst,
                                const float* __restrict__ att,
                                float* __restrict__ elog, float* __restrict__ mmax) {
  int e = blockIdx.x * 256 + threadIdx.x;
  if (e >= kE) return;
  int s = src[e], d = dst[e];
  const float4* pl = (const float4*)(xl + s * kHC);
  const float4* pr = (const float4*)(xr + d * kHC);
  const float4* pa = (const float4*)att;
  float acc0 = 0.0f, acc1 = 0.0f;
#pragma unroll
  for (int i = 0; i < 16; ++i) {        // head 0: channels 0..63
    float4 l = pl[i], r = pr[i], a = pa[i];
    float z;
    z = l.x + r.x; z = z > 0.0f ? z : 0.2f * z; acc0 += z * a.x;
    z = l.y + r.y; z = z > 0.0f ? z : 0.2f * z; acc0 += z * a.y;
    z = l.z + r.z; z = z > 0.0f ? z : 0.2f * z; acc0 += z * a.z;
    z = l.w + r.w; z = z > 0.0f ? z : 0.2f * z; acc0 += z * a.w;
  }
#pragma unroll
  for (int i = 16; i < 32; ++i) {       // head 1: channels 64..127
    float4 l = pl[i], r = pr[i], a = pa[i];
    float z;
    z = l.x + r.x; z = z > 0.0f ? z : 0.2f * z; acc1 += z * a.x;
    z = l.y + r.y; z = z > 0.0f ? z : 0.2f * z; acc1 += z * a.y;
    z = l.z + r.z; z = z > 0.0f ? z : 0.2f * z; acc1 += z * a.z;
    z = l.w + r.w; z = z > 0.0f ? z : 0.2f * z; acc1 += z * a.w;
  }
  elog[2 * e]     = acc0;
  elog[2 * e + 1] = acc1;
  atomicMaxF32(&mmax[2 * d],     acc0);
  atomicMaxF32(&mmax[2 * d + 1], acc1);
}

// pass 2: a = exp(e - max[dst]); segment sum of a
__global__ void edge_exp_sum(float* __restrict__ elog, const float* __restrict__ mmax,
                             const int* __restrict__ dst, float* __restrict__ dsum) {
  int idx = blockIdx.x * 256 + threadIdx.x;
  if (idx >= kE * 2) return;
  int e = idx >> 1, h = idx & 1;
  int d = dst[e];
  float a = expf(elog[idx] - mmax[2 * d + h]);
  elog[idx] = a;
  atomicAdd(&dsum[2 * d + h], a);
}

// pass 3: acc[dst] += xl[src] * alpha   (one thread per edge x 4-channel group)
__global__ void edge_scatter(const float* __restrict__ xl, const float* __restrict__ elog,
                             const float* __restrict__ dsum,
                             const int* __restrict__ src, const int* __restrict__ dst,
                             float* __restrict__ acc) {
  int idx = blockIdx.x * 256 + threadIdx.x;
  if (idx >= kE * 32) return;
  int e = idx >> 5, g = idx & 31;       // g: float4 group within 128 channels
  int h = g >> 4;                       // head = group/16
  int s = src[e], d = dst[e];
  float alpha = elog[2 * e + h] / (dsum[2 * d + h] + 1e-16f);
  float4 v = ((const float4*)(xl + s * kHC))[g];
  float* o = acc + d * kHC + g * 4;
  atomicAdd(o + 0, v.x * alpha);
  atomicAdd(o + 1, v.y * alpha);
  atomicAdd(o + 2, v.z * alpha);
  atomicAdd(o + 3, v.w * alpha);
}

// mean over heads + bc, write or accumulate into out[n_dst, 64]
__global__ void gat_finalize(const float* __restrict__ acc, const float* __restrict__ bc,
                             float* __restrict__ out, int n_dst, int accumulate) {
  int idx = blockIdx.x * 256 + threadIdx.x;
  if (idx >= n_dst * kH) return;
  int d = idx / kH, c = idx % kH;
  float v = 0.5f * (acc[d * kHC + c] + acc[d * kHC + kH + c]) + bc[c];
  if (accumulate) out[idx] += v;
  else            out[idx] = v;
}

// out = concat(xm, xd) @ Wi + bi   (tiny: 2000x5 threads)
__global__ void final_linear(const float* __restrict__ xm, const float* __restrict__ xd,
                             const float* __restrict__ Wi, const float* __restrict__ bi,
                             float* __restrict__ out) {
  int idx = blockIdx.x * 256 + threadIdx.x;
  if (idx >= kSamp * kNC) return;
  int s = idx / kNC, c = idx % kNC;
  float a = bi[c];
  const float* pm = xm + s * kH;
  const float* pd = xd + s * kH;
#pragma unroll 8
  for (int k = 0; k < kH; ++k) a += pm[k] * Wi[k * kNC + c];
#pragma unroll 8
  for (int k = 0; k < kH; ++k) a += pd[k] * Wi[(kH + k) * kNC + c];
  out[idx] = a;
}

// ---------------------------------------------------------------------------
// Host-side orchestration
// ---------------------------------------------------------------------------
struct Scr { float *xl, *xr, *elog, *mmax, *dsum, *acc; };

static void launch_gemm(hipStream_t st, const float* A, const float* B, const float* bias,
                        float* C, int M, int N, int K, int flags) {
  constexpr int MBLK = 4;
  int ntn = N / 16;                               // N % 16 == 0 for all calls
  int msets = (M + 16 * MBLK - 1) / (16 * MBLK);
  int total = msets * ntn;
  dim3 blk(32, 4, 1);
  int grid = (total + 3) / 4;
  if (flags == GF_BIAS) {
    gemm_wmma<GF_BIAS, MBLK><<<grid, blk, 0, st>>>(A, B, bias, C, M, N, K, total, ntn);
  } else if (flags == (GF_BIAS | GF_RELU)) {
    gemm_wmma<GF_BIAS | GF_RELU, MBLK><<<grid, blk, 0, st>>>(A, B, bias, C, M, N, K, total, ntn);
  } else {
    gemm_wmma<GF_ACC | GF_BIAS | GF_RELU, MBLK><<<grid, blk, 0, st>>>(A, B, bias, C, M, N, K, total, ntn);
  }
}

static void gat_rel(hipStream_t st,
                    const float* x_src, int n_src, const float* x_dst, int n_dst,
                    const float* Wl, const float* bl, const float* Wr, const float* br,
                    const float* att, const float* bc,
                    const int* src, const int* dst,
                    float* out, bool accumulate, const Scr& s) {
  launch_gemm(st, x_src, Wl, bl, s.xl, n_src, kHC, kP, GF_BIAS);
  launch_gemm(st, x_dst, Wr, br, s.xr, n_dst, kHC, kP, GF_BIAS);
  int nh = n_dst * 2;
  fill_f32<<<(nh + 255) / 256, 256, 0, st>>>(s.mmax, -__builtin_inff(), nh);
  fill_f32<<<(nh + 255) / 256, 256, 0, st>>>(s.dsum, 0.0f, nh);
  int na = n_dst * kHC;
  fill_f32<<<(na + 255) / 256, 256, 0, st>>>(s.acc, 0.0f, na);
  edge_logits_max<<<(kE + 255) / 256, 256, 0, st>>>(s.xl, s.xr, src, dst, att, s.elog, s.mmax);
  edge_exp_sum<<<(kE * 2 + 255) / 256, 256, 0, st>>>(s.elog, s.mmax, dst, s.dsum);
  edge_scatter<<<(kE * 32 + 255) / 256, 256, 0, st>>>(s.xl, s.elog, s.dsum, src, dst, s.acc);
  gat_finalize<<<(n_dst * kH + 255) / 256, 256, 0, st>>>(s.acc, bc, out, n_dst, accumulate ? 1 : 0);
}

static void run_layer(hipStream_t st,
                      const float* xm, const float* xd, const float* xg,
                      float* om, float* od, float* og,
                      const float* Wl, const float* bl, const float* Wr, const float* br,
                      const float* att, const float* bc, const float* Ws, const float* bs,
                      const int* const* esrc, const int* const* edst, const Scr& s) {
  const int WLS = kP * kHC, BLS = kHC, ATS = kHC, BCS = kH, WSS = kP * kH, BSS = kH;
  // relation order: 0=mrna->genes, 1=genes->mrna, 2=dnam->genes, 3=genes->dnam
  gat_rel(st, xm, kSamp,  xg, kGenes, Wl + 0*WLS, bl + 0*BLS, Wr + 0*WLS, br + 0*BLS,
          att + 0*ATS, bc + 0*BCS, esrc[0], edst[0], og, false, s);
  gat_rel(st, xd, kSamp,  xg, kGenes, Wl + 2*WLS, bl + 2*BLS, Wr + 2*WLS, br + 2*BLS,
          att + 2*ATS, bc + 2*BCS, esrc[2], edst[2], og, true, s);
  gat_rel(st, xg, kGenes, xm, kSamp,  Wl + 1*WLS, bl + 1*BLS, Wr + 1*WLS, br + 1*BLS,
          att + 1*ATS, bc + 1*BCS, esrc[1], edst[1], om, false, s);
  gat_rel(st, xg, kGenes, xd, kSamp,  Wl + 3*WLS, bl + 3*BLS, Wr + 3*WLS, br + 3*BLS,
          att + 3*ATS, bc + 3*BCS, esrc[3], edst[3], od, false, s);
  // self-loop: h = relu(h + x @ Ws + bs)   (fused into WMMA GEMM with ACC)
  launch_gemm(st, xm, Ws + 0*WSS, bs + 0*BSS, om, kSamp,  kH, kP, GF_ACC | GF_BIAS | GF_RELU);
  launch_gemm(st, xd, Ws + 1*WSS, bs + 1*BSS, od, kSamp,  kH, kP, GF_ACC | GF_BIAS | GF_RELU);
  launch_gemm(st, xg, Ws + 2*WSS, bs + 2*BSS, og, kGenes, kH, kP, GF_ACC | GF_BIAS | GF_RELU);
}

extern "C" void kernel_launch(void* const* d_in, const int* in_sizes, int n_in,
                              void* d_out, int out_size, void* d_ws, size_t ws_size,
                              hipStream_t stream) {
  const float* x_mrna  = (const float*)d_in[0];
  const float* x_dnam  = (const float*)d_in[1];
  const float* x_genes = (const float*)d_in[2];
  const float* Wp_mrna = (const float*)d_in[3];
  const float* bp_mrna = (const float*)d_in[4];
  const float* Wp_dnam = (const float*)d_in[5];
  const float* bp_dnam = (const float*)d_in[6];
  const float* Wl1 = (const float*)d_in[7];
  const float* bl1 = (const float*)d_in[8];
  const float* Wr1 = (const float*)d_in[9];
  const float* br1 = (const float*)d_in[10];
  const float* att1 = (const float*)d_in[11];
  const float* bc1 = (const float*)d_in[12];
  const float* Wl2 = (const float*)d_in[13];
  const float* bl2 = (const float*)d_in[14];
  const float* Wr2 = (const float*)d_in[15];
  const float* br2 = (const float*)d_in[16];
  const float* att2 = (const float*)d_in[17];
  const float* bc2 = (const float*)d_in[18];
  const float* Ws1 = (const float*)d_in[19];
  const float* bs1 = (const float*)d_in[20];
  const float* Ws2 = (const float*)d_in[21];
  const float* bs2 = (const float*)d_in[22];
  const float* Wi  = (const float*)d_in[23];
  const float* bi  = (const float*)d_in[24];
  const int* esrc[4] = {(const int*)d_in[25], (const int*)d_in[27],
                        (const int*)d_in[29], (const int*)d_in[31]};
  const int* edst[4] = {(const int*)d_in[26], (const int*)d_in[28],
                        (const int*)d_in[30], (const int*)d_in[32]};

  // bump-allocate workspace (~25 MB of fp32)
  float* p = (float*)d_ws;
  auto take = [&](size_t n) { float* r = p; p += n; return r; };
  float* xm_a = take((size_t)kSamp * kP);
  float* xd_a = take((size_t)kSamp * kP);
  float* xm_b = take((size_t)kSamp * kP);
  float* xd_b = take((size_t)kSamp * kP);
  float* xg_a = take((size_t)kGenes * kP);
  float* xg_b = take((size_t)kGenes * kP);
  Scr s;
  s.xl   = take((size_t)kGenes * kHC);
  s.xr   = take((size_t)kGenes * kHC);
  s.elog = take((size_t)kE * 2);
  s.mmax = take((size_t)kGenes * 2);
  s.dsum = take((size_t)kGenes * 2);
  s.acc  = take((size_t)kGenes * kHC);

  // input projections (the big K=2000 GEMMs): x = relu(X @ Wp + bp)
  launch_gemm(stream, x_mrna, Wp_mrna, bp_mrna, xm_a, kSamp, kP, kInDim, GF_BIAS | GF_RELU);
  launch_gemm(stream, x_dnam, Wp_dnam, bp_dnam, xd_a, kSamp, kP, kInDim, GF_BIAS | GF_RELU);

  run_layer(stream, xm_a, xd_a, x_genes, xm_b, xd_b, xg_a,
            Wl1, bl1, Wr1, br1, att1, bc1, Ws1, bs1, esrc, edst, s);
  run_layer(stream, xm_b, xd_b, xg_a, xm_a, xd_a, xg_b,
            Wl2, bl2, Wr2, br2, att2, bc2, Ws2, bs2, esrc, edst, s);

  final_linear<<<(kSamp * kNC + 255) / 256, 256, 0, stream>>>(
      xm_a, xd_a, Wi, bi, (float*)d_out);
}